// BilinearMixture_11424613008076
// MI455X (gfx1250) — compile-verified
//
#include <hip/hip_runtime.h>
#include <hip/hip_bf16.h>

typedef __attribute__((ext_vector_type(2))) float v2f;
typedef __attribute__((ext_vector_type(4))) float v4f;
typedef __attribute__((ext_vector_type(8))) float v8f;

#define WAVES_PER_BLOCK 4
#define P_STRIDE 68   // 64 + 4 pad: bank-conflict-free v2f A-fragment reads
#define D_STRIDE 17

// ---------------------------------------------------------------------------
// Kernel 1: collapse basis_weights [K,64] and weights_scalars [K,C] into one
// combined weight, stored TRANSPOSED: Wt[n][d] = sum_k bw[k,d]*ws[k,c=n],
// padded to 16 columns (n=C..15 are zero). 16*64 floats in workspace.
// ---------------------------------------------------------------------------
__global__ void bilinear_prep_W(const float* __restrict__ bw,
                                const float* __restrict__ ws,
                                float* __restrict__ Wt, int K, int C) {
    int tid = threadIdx.x;          // 1024 threads: 16 cols x 64 d
    int n = tid >> 6;
    int d = tid & 63;
    float s = 0.0f;
    if (n < C) {
        for (int k = 0; k < K; ++k)
            s += bw[k * 64 + d] * ws[k * C + n];
    }
    Wt[n * 64 + d] = s;
}

// ---------------------------------------------------------------------------
// Kernel 2: one wave32 per 16-edge tile.
//   phase 1: coalesced gather of u/v rows (lane pair -> one row, b128 loads),
//            compute p = u*v elementwise, stage in LDS (stride-68 rows).
//   phase 2: 16x V_WMMA_F32_16X16X4_F32 accumulating over D=64.
//   phase 3: D -> LDS, lanes 0..15 each own one edge: + biases, softmax, store.
// ---------------------------------------------------------------------------
__global__ void __launch_bounds__(32 * WAVES_PER_BLOCK)
bilinear_mixture_kernel(const float* __restrict__ u_feat,
                        const float* __restrict__ v_feat,
                        const int*   __restrict__ u_idx,
                        const int*   __restrict__ v_idx,
                        const float* __restrict__ Wt,     // [16][64]
                        const float* __restrict__ ubias,  // [NU][C]
                        const float* __restrict__ ibias,  // [NI][C]
                        float*       __restrict__ out,    // [E][C]
                        int E, int C) {
    __shared__ __align__(16) float sP[WAVES_PER_BLOCK][16 * P_STRIDE];
    __shared__ __align__(16) float sD[WAVES_PER_BLOCK][16 * D_STRIDE];

    const int lane = threadIdx.x & 31;
    const int w    = threadIdx.x >> 5;
    const long tileBase = ((long)blockIdx.x * WAVES_PER_BLOCK + w) * 16;

    // ---- phase 1: gather + elementwise product into LDS -------------------
    {
        const int r    = lane >> 1;        // row (edge within tile) 0..15
        const int half = lane & 1;         // which 32-float half of the row
        long eRow = tileBase + r;
        if (eRow >= E) eRow = E - 1;       // clamp: keeps EXEC uniform
        const int ui = u_idx[eRow];
        const int vi = v_idx[eRow];
        const v4f* __restrict__ up =
            (const v4f*)(u_feat + (long)ui * 64 + half * 32);
        const v4f* __restrict__ vp =
            (const v4f*)(v_feat + (long)vi * 64 + half * 32);
        float* pRow = &sP[w][r * P_STRIDE + half * 32];
#pragma unroll
        for (int i = 0; i < 8; ++i) {
            v4f a = up[i];                 // global_load_b128
            v4f b = vp[i];
            *(v4f*)&pRow[4 * i] = a * b;   // ds_store_b128
        }
    }
    __syncthreads();

    // ---- phase 2: [16,64] x [64,16] via 16x v_wmma_f32_16x16x4_f32 --------
    const int m = lane & 15;               // A row / B col / D col
    const int h = lane >> 4;               // K sub-pair selector
    const float* pW = &sP[w][0];
    v8f acc = {};
#pragma unroll
    for (int t = 0; t < 16; ++t) {
        const int d0 = 4 * t + 2 * h;
        v2f a = *(const v2f*)&pW[m * P_STRIDE + d0];  // A: P[m][d0..d0+1]
        v2f b = *(const v2f*)&Wt[m * 64 + d0];        // B: W[d0..d0+1][n=m]
        acc = __builtin_amdgcn_wmma_f32_16x16x4_f32(
            /*neg_a=*/false, a, /*neg_b=*/false, b,
            /*c_mod=*/(short)0, acc, /*reuse_a=*/false, /*reuse_b=*/false);
    }

    // ---- phase 3: D -> LDS, then per-edge bias + softmax ------------------
#pragma unroll
    for (int r2 = 0; r2 < 8; ++r2)
        sD[w][(r2 + 8 * h) * D_STRIDE + m] = acc[r2];
    __syncthreads();

    if (lane < 16) {
        const long e = tileBase + lane;
        if (e < E) {
            const int ui = u_idx[e];
            const int vi = v_idx[e];
            float vals[16];
            float mx = -3.4e38f;
#pragma unroll 5
            for (int c = 0; c < C; ++c) {
                float x = sD[w][lane * D_STRIDE + c]
                        + ubias[(long)ui * C + c]
                        + ibias[(long)vi * C + c];
                vals[c] = x;
                mx = fmaxf(mx, x);
            }
            float sum = 0.0f;
#pragma unroll 5
            for (int c = 0; c < C; ++c) {
                float ex = __expf(vals[c] - mx);   // v_exp_f32
                vals[c] = ex;
                sum += ex;
            }
            const float inv = __frcp_rn(sum);
            float* o = out + e * (long)C;
#pragma unroll 5
            for (int c = 0; c < C; ++c)
                o[c] = vals[c] * inv;
        }
    }
}

// ---------------------------------------------------------------------------
extern "C" void kernel_launch(void* const* d_in, const int* in_sizes, int n_in,
                              void* d_out, int out_size, void* d_ws, size_t ws_size,
                              hipStream_t stream) {
    const float* u_feat = (const float*)d_in[0];   // [NU,64]
    const float* v_feat = (const float*)d_in[1];   // [NI,64]
    const int*   u_idx  = (const int*)  d_in[2];   // [E]
    const int*   v_idx  = (const int*)  d_in[3];   // [E]
    const float* bw     = (const float*)d_in[4];   // [K,64]
    const float* ws     = (const float*)d_in[5];   // [K,C]
    const float* ubias  = (const float*)d_in[6];   // [NU,C]
    const float* ibias  = (const float*)d_in[7];   // [NI,C]
    float*       out    = (float*)d_out;

    const int E = in_sizes[2];
    const int K = in_sizes[4] / 64;
    const int C = in_sizes[5] / (K > 0 ? K : 1);   // 5

    float* Wt = (float*)d_ws;                      // 16*64 floats scratch

    bilinear_prep_W<<<1, 1024, 0, stream>>>(bw, ws, Wt, K, C);

    const int tiles  = (E + 15) / 16;
    const int blocks = (tiles + WAVES_PER_BLOCK - 1) / WAVES_PER_BLOCK;
    bilinear_mixture_kernel<<<blocks, 32 * WAVES_PER_BLOCK, 0, stream>>>(
        u_feat, v_feat, u_idx, v_idx, Wt, ubias, ibias, out, E, C);
}